// DocREModel_29145648071160
// MI455X (gfx1250) — compile-verified
//
#include <hip/hip_runtime.h>
#include <math.h>

// ---------------------------------------------------------------------------
// Problem constants (from reference)
// ---------------------------------------------------------------------------
#define BB   4
#define CC   512
#define HH   768
#define NHD  12
#define NE   20
#define MMN  23
#define NP   380
#define QSZ  128
#define POSZ 128
#define EMB  768
#define BLK  64
#define NBK  12
#define NCLS 97
#define KBL  (EMB*BLK)      // 49152
#define NEGV (-10000.0f)
#define INV_SQRT_EMB 0.036084391824351614f   // 1/sqrt(768)

typedef float v2f __attribute__((ext_vector_type(2)));
typedef float v8f __attribute__((ext_vector_type(8)));

// ---------------------------------------------------------------------------
// Generic fp32 WMMA GEMM:  C = act( alpha * (A @ B) + bias + resid )
//   A composed of up to 3 row-segments (concat), each with optional per-row
//   gather index.  Optional transposed B.  Optional batching (blockIdx.z).
//   Block: 256 thr = 8 waves -> 64x64 output tile.  Wave grid 4(row)x2(col),
//   each wave owns 16x32 (two 16x16 accumulators sharing the A fragment).
//   K staged in LDS panels of 32 -> 16 WMMAs per wave between barriers.
// ---------------------------------------------------------------------------
__global__ __launch_bounds__(256)
void gemm_wmma_f32(const float* __restrict__ A0, const int* __restrict__ A0i, int lda0, int w0,
                   const float* __restrict__ A1, const int* __restrict__ A1i, int lda1, int w1,
                   const float* __restrict__ A2, const int* __restrict__ A2i, int lda2, int w2,
                   const float* __restrict__ Bm, int ldb, int transB,
                   const float* __restrict__ bias,
                   const float* __restrict__ resid, int ldres,
                   float* __restrict__ C, int ldc,
                   int M, int N, int K, float alpha, int act,
                   long sA0, long sB, long sC, long sRes)
{
    __shared__ float As[64][33];   // 64 rows x 32 k  (+pad, stride 33 odd)
    __shared__ float Bs[32][65];   // 32 k  x 64 cols (+pad, stride 65 odd)

    const int tid   = threadIdx.x;
    const int batch = blockIdx.z;
    const float* a0 = A0 + (long)batch * sA0;
    const float* bp = Bm + (long)batch * sB;
    float*       cp = C  + (long)batch * sC;
    const float* rp = resid ? (resid + (long)batch * sRes) : (const float*)0;

    const int rowBase = blockIdx.y * 64;
    const int colBase = blockIdx.x * 64;

    const int lane = tid & 31;
    const int wave = tid >> 5;
    const int half = (lane >> 4) & 1;
    const int l15  = lane & 15;
    const int wr   = wave & 3;   // 0..3 : 16-row sub-tile
    const int wcg  = wave >> 2;  // 0..1 : 32-col group

    v8f acc0 = {};
    v8f acc1 = {};

    for (int k0 = 0; k0 < K; k0 += 32) {
        // ---- stage A (64 x 32), 8 elems/thread ----
        #pragma unroll
        for (int it = 0; it < 8; ++it) {
            const int t  = tid + it * 256;
            const int r  = t >> 5, kk = t & 31;
            const int gr = rowBase + r, gk = k0 + kk;
            float v = 0.f;
            if (gr < M && gk < K) {
                int kx = gk;
                if (kx < w0) {
                    int rr = A0i ? A0i[gr] : gr;
                    v = a0[(long)rr * lda0 + kx];
                } else if ((kx -= w0) < w1) {
                    int rr = A1i ? A1i[gr] : gr;
                    v = A1[(long)rr * lda1 + kx];
                } else {
                    kx -= w1;
                    int rr = A2i ? A2i[gr] : gr;
                    v = A2[(long)rr * lda2 + kx];
                }
            }
            As[r][kk] = v;
        }
        // ---- stage B (32 x 64), 8 elems/thread ----
        #pragma unroll
        for (int it = 0; it < 8; ++it) {
            const int t  = tid + it * 256;
            const int kk = t >> 6, c = t & 63;
            const int gk = k0 + kk, gc = colBase + c;
            float v = 0.f;
            if (gk < K && gc < N)
                v = transB ? bp[(long)gc * ldb + gk] : bp[(long)gk * ldb + gc];
            Bs[kk][c] = v;
        }
        __syncthreads();

        // ---- 16 x WMMA f32 16x16x4 per wave per panel ----
        #pragma unroll
        for (int kk = 0; kk < 8; ++kk) {
            const int kb = kk * 4 + half * 2;   // lanes0-15: K=0,1 ; lanes16-31: K=2,3
            v2f av, b0, b1;
            av.x = As[wr * 16 + l15][kb];
            av.y = As[wr * 16 + l15][kb + 1];
            b0.x = Bs[kb][wcg * 32 + l15];
            b0.y = Bs[kb + 1][wcg * 32 + l15];
            b1.x = Bs[kb][wcg * 32 + 16 + l15];
            b1.y = Bs[kb + 1][wcg * 32 + 16 + l15];
            acc0 = __builtin_amdgcn_wmma_f32_16x16x4_f32(false, av, false, b0,
                                                         (short)0, acc0, false, false);
            acc1 = __builtin_amdgcn_wmma_f32_16x16x4_f32(false, av, false, b1,
                                                         (short)0, acc1, false, false);
        }
        __syncthreads();
    }

    // ---- epilogue: C/D layout VGPR r -> M = r + 8*half, N = l15 ----
    #pragma unroll
    for (int sub = 0; sub < 2; ++sub) {
        const v8f acc = sub ? acc1 : acc0;
        const int col = colBase + wcg * 32 + sub * 16 + l15;
        if (col < N) {
            const float bval = bias ? bias[col] : 0.f;
            #pragma unroll
            for (int r = 0; r < 8; ++r) {
                const int row = rowBase + wr * 16 + r + half * 8;
                if (row < M) {
                    float v = acc[r] * alpha + bval;
                    if (rp) v += rp[(long)row * ldres + col];
                    if (act == 1) v = tanhf(v);
                    cp[(long)row * ldc + col] = v;
                }
            }
        }
    }
}

// ---------------------------------------------------------------------------
// Bilinear GEMM: out[p,d] = sum_{nb,i,j} hsE[p,nb*64+i]*tsE[p,nb*64+j]*Wbl[(nb,i,j),d]
// 49152-wide A generated on the fly in LDS (never materialized in HBM).
// Same 64x64 / K-panel-32 tiling as the generic kernel.
// ---------------------------------------------------------------------------
__global__ __launch_bounds__(256)
void gemm_bilinear_wmma(const float* __restrict__ H1, const float* __restrict__ H2,
                        const float* __restrict__ Wbl, const float* __restrict__ bias,
                        float* __restrict__ C, int M)
{
    __shared__ float As[64][33];
    __shared__ float Bs[32][65];

    const int tid     = threadIdx.x;
    const int rowBase = blockIdx.y * 64;
    const int colBase = blockIdx.x * 64;

    const int lane = tid & 31;
    const int wave = tid >> 5;
    const int half = (lane >> 4) & 1;
    const int l15  = lane & 15;
    const int wr   = wave & 3;
    const int wcg  = wave >> 2;

    v8f acc0 = {};
    v8f acc1 = {};

    for (int k0 = 0; k0 < KBL; k0 += 32) {
        #pragma unroll
        for (int it = 0; it < 8; ++it) {
            const int t  = tid + it * 256;
            const int r  = t >> 5, kk = t & 31;
            const int gr = rowBase + r;
            const int k  = k0 + kk;
            float v = 0.f;
            if (gr < M) {
                const int nb = k >> 12;          // /4096
                const int i  = (k >> 6) & 63;
                const int j  = k & 63;
                v = H1[(long)gr * EMB + (nb << 6) + i] *
                    H2[(long)gr * EMB + (nb << 6) + j];
            }
            As[r][kk] = v;
        }
        #pragma unroll
        for (int it = 0; it < 8; ++it) {
            const int t  = tid + it * 256;
            const int kk = t >> 6, c = t & 63;
            Bs[kk][c] = Wbl[(long)(k0 + kk) * EMB + colBase + c];
        }
        __syncthreads();
        #pragma unroll
        for (int kk = 0; kk < 8; ++kk) {
            const int kb = kk * 4 + half * 2;
            v2f av, b0, b1;
            av.x = As[wr * 16 + l15][kb];
            av.y = As[wr * 16 + l15][kb + 1];
            b0.x = Bs[kb][wcg * 32 + l15];
            b0.y = Bs[kb + 1][wcg * 32 + l15];
            b1.x = Bs[kb][wcg * 32 + 16 + l15];
            b1.y = Bs[kb + 1][wcg * 32 + 16 + l15];
            acc0 = __builtin_amdgcn_wmma_f32_16x16x4_f32(false, av, false, b0,
                                                         (short)0, acc0, false, false);
            acc1 = __builtin_amdgcn_wmma_f32_16x16x4_f32(false, av, false, b1,
                                                         (short)0, acc1, false, false);
        }
        __syncthreads();
    }

    #pragma unroll
    for (int sub = 0; sub < 2; ++sub) {
        const v8f acc = sub ? acc1 : acc0;
        const int col = colBase + wcg * 32 + sub * 16 + l15;   // col < 768 by grid
        const float bval = bias[col];
        #pragma unroll
        for (int r = 0; r < 8; ++r) {
            const int row = rowBase + wr * 16 + r + half * 8;
            if (row < M)
                C[(long)row * EMB + col] = acc[r] + bval;
        }
    }
}

// ---------------------------------------------------------------------------
// Small support kernels
// ---------------------------------------------------------------------------
__global__ void k_idx(const int* __restrict__ hts, int* __restrict__ hif,
                      int* __restrict__ tif, int n)
{
    int i = blockIdx.x * blockDim.x + threadIdx.x;
    if (i < n) { hif[i] = hts[2 * i]; tif[i] = hts[2 * i + 1]; }
}

// entm[b,e,m,:] = seq[b, midx[b,e,m], :] * mmask[b,e,m]
__global__ __launch_bounds__(256)
void k_entm(const float* __restrict__ seq, const int* __restrict__ midx,
            const float* __restrict__ mmask, float* __restrict__ entm)
{
    const int bem = blockIdx.x;            // (b*NE+e)*MM + m
    const int b   = bem / (NE * MMN);
    const float mk = mmask[bem];
    const int idx  = midx[bem];
    const float* src = seq + ((long)b * CC + idx) * HH;
    float* dst = entm + (long)bem * HH;
    for (int h = threadIdx.x; h < HH; h += 256) dst[h] = src[h] * mk;
}

// eatt[b,e,hd,:] = (sum_m mask * att[b,hd,midx,:]) / cnt
__global__ __launch_bounds__(256)
void k_entatt(const float* __restrict__ att, const int* __restrict__ midx,
              const float* __restrict__ mmask, float* __restrict__ eatt)
{
    const int blk = blockIdx.x;            // (b*NE+e)*NHD + hd
    const int hd  = blk % NHD;
    const int be  = blk / NHD;
    const int b   = be / NE;
    const int tid = threadIdx.x;
    __shared__ float smask[MMN];
    __shared__ int   sidx[MMN];
    __shared__ float scnt;
    if (tid < MMN) { smask[tid] = mmask[be * MMN + tid]; sidx[tid] = midx[be * MMN + tid]; }
    __syncthreads();
    if (tid == 0) { float c = 0.f; for (int m = 0; m < MMN; ++m) c += smask[m]; scnt = c; }
    __syncthreads();
    const float inv = 1.f / scnt;
    const float* ab = att + ((long)b * NHD + hd) * CC * CC;
    float* dst = eatt + (long)blk * CC;
    for (int c = tid; c < CC; c += 256) {
        float s = 0.f;
        for (int m = 0; m < MMN; ++m)
            if (smask[m] > 0.f) s += ab[(long)sidx[m] * CC + c];
        dst[c] = s * inv;
    }
}

// htatt[b,p,:] = normalize( mean_hd eatt[hi]*eatt[ti] )
__global__ __launch_bounds__(256)
void k_htatt(const float* __restrict__ eatt, const int* __restrict__ hif,
             const int* __restrict__ tif, float* __restrict__ htatt)
{
    const int bp  = blockIdx.x;
    const int b   = bp / NP;
    const int tid = threadIdx.x;
    __shared__ float vals[CC];
    __shared__ float red[256];
    const int hi = hif[bp], ti = tif[bp];
    const float* ph = eatt + (long)(b * NE + hi) * NHD * CC;
    const float* pt = eatt + (long)(b * NE + ti) * NHD * CC;
    float local = 0.f;
    for (int c = tid; c < CC; c += 256) {
        float s = 0.f;
        for (int hd = 0; hd < NHD; ++hd) s += ph[hd * CC + c] * pt[hd * CC + c];
        s *= (1.f / (float)NHD);
        vals[c] = s;
        local += s;
    }
    red[tid] = local; __syncthreads();
    for (int off = 128; off > 0; off >>= 1) {
        if (tid < off) red[tid] += red[tid + off];
        __syncthreads();
    }
    const float inv = 1.f / (red[0] + 1e-5f);
    for (int c = tid; c < CC; c += 256) htatt[(long)bp * CC + c] = vals[c] * inv;
}

// mention attention: G = htq . eq / sqrt(EMB), mask, softmax over 23, pool entm
__global__ __launch_bounds__(256)
void k_mention(const float* __restrict__ htq, const float* __restrict__ eq,
               const float* __restrict__ entm, const float* __restrict__ mmask,
               const int* __restrict__ hif, const int* __restrict__ tif,
               float* __restrict__ hs, float* __restrict__ ts)
{
    const int bp  = blockIdx.x;
    const int b   = bp / NP;
    const int tid = threadIdx.x;
    __shared__ float qv[QSZ];
    __shared__ float g[MMN];
    __shared__ float sinv;
    if (tid < QSZ) qv[tid] = htq[(long)bp * QSZ + tid];
    __syncthreads();
    for (int side = 0; side < 2; ++side) {
        const int e = side ? tif[bp] : hif[bp];
        const long erow = (long)(b * NE + e) * MMN;
        if (tid < MMN) {
            const float* ep = eq + (erow + tid) * QSZ;
            float d = 0.f;
            for (int q = 0; q < QSZ; ++q) d += qv[q] * ep[q];
            d *= INV_SQRT_EMB;
            if (!(mmask[erow + tid] > 0.f)) d = NEGV;
            g[tid] = d;
        }
        __syncthreads();
        if (tid == 0) {
            float mx = g[0];
            for (int m = 1; m < MMN; ++m) mx = fmaxf(mx, g[m]);
            float s = 0.f;
            for (int m = 0; m < MMN; ++m) { float e2 = expf(g[m] - mx); g[m] = e2; s += e2; }
            sinv = 1.f / s;
        }
        __syncthreads();
        const float inv = sinv;
        float* dst = (side ? ts : hs) + (long)bp * HH;
        const float* em = entm + erow * HH;
        for (int h = tid; h < HH; h += 256) {
            float a = 0.f;
            for (int m = 0; m < MMN; ++m) a += g[m] * em[(long)m * HH + h];
            dst[h] = a * inv;
        }
        __syncthreads();
    }
}

// masked softmax over 380 pair-graph scores, in place
__global__ __launch_bounds__(256)
void k_pairsoftmax(float* __restrict__ sc, const float* __restrict__ vis)
{
    const int bp  = blockIdx.x;
    const int tid = threadIdx.x;
    float* row = sc + (long)bp * NP;
    const float* vm = vis + (long)bp * NP;
    __shared__ float red[256];
    float mx = -3.0e38f;
    for (int q = tid; q < NP; q += 256) {
        float v = row[q];
        if (!(vm[q] > 0.f)) v = NEGV;
        row[q] = v;
        mx = fmaxf(mx, v);
    }
    red[tid] = mx; __syncthreads();
    for (int off = 128; off > 0; off >>= 1) {
        if (tid < off) red[tid] = fmaxf(red[tid], red[tid + off]);
        __syncthreads();
    }
    mx = red[0];
    __syncthreads();
    float s = 0.f;
    for (int q = tid; q < NP; q += 256) {
        float e = expf(row[q] - mx);
        row[q] = e; s += e;
    }
    red[tid] = s; __syncthreads();
    for (int off = 128; off > 0; off >>= 1) {
        if (tid < off) red[tid] += red[tid + off];
        __syncthreads();
    }
    const float inv = 1.f / red[0];
    for (int q = tid; q < NP; q += 256) row[q] *= inv;
}

// ---------------------------------------------------------------------------
// Host orchestration
// ---------------------------------------------------------------------------
extern "C" void kernel_launch(void* const* d_in, const int* in_sizes, int n_in,
                              void* d_out, int out_size, void* d_ws, size_t ws_size,
                              hipStream_t stream)
{
    (void)in_sizes; (void)n_in; (void)out_size; (void)ws_size;

    const float* seq   = (const float*)d_in[0];
    const float* att   = (const float*)d_in[1];
    const int*   midx  = (const int*)  d_in[2];
    const float* mmask = (const float*)d_in[3];
    const int*   hts   = (const int*)  d_in[4];
    const float* vis   = (const float*)d_in[5];
    const float* Wcq = (const float*)d_in[6],  *bcq = (const float*)d_in[7];
    const float* Weq = (const float*)d_in[8],  *beq = (const float*)d_in[9];
    const float* Whe = (const float*)d_in[10], *bhe = (const float*)d_in[11];
    const float* Wte = (const float*)d_in[12], *bte = (const float*)d_in[13];
    const float* Wbl = (const float*)d_in[14], *bbl = (const float*)d_in[15];
    const float* Wpp = (const float*)d_in[16], *bpp = (const float*)d_in[17];
    const float* Wgq = (const float*)d_in[18];
    const float* Wgk = (const float*)d_in[19];
    const float* Wgv = (const float*)d_in[20];
    const float* Wp1 = (const float*)d_in[21], *bp1 = (const float*)d_in[22];
    const float* Wp2 = (const float*)d_in[23], *bp2 = (const float*)d_in[24];
    const float* pos = (const float*)d_in[25];
    float* out = (float*)d_out;

    // ---- workspace layout (floats), with lifetime-based aliasing ----
    float* ws = (float*)d_ws;
    const long ENTM_O  = 0;                          // 4*20*23*768 = 1,413,120  (-> SC later)
    const long EQ_O    = ENTM_O  + 1413120;          // 4*20*23*128 =   235,520
    const long EATT_O  = EQ_O    + 235520;           // 4*20*12*512 =   491,520
    const long HTATT_O = EATT_O  + 491520;           // 4*380*512   =   778,240  (-> HTQ later)
    const long RS_O    = HTATT_O + 778240;           // 4*380*768   = 1,167,360  (-> VB later)
    const long HS_O    = RS_O    + 1167360;          //               1,167,360  (-> QB later)
    const long TS_O    = HS_O    + 1167360;          //               1,167,360  (-> KB later)
    const long HSE_O   = TS_O    + 1167360;
    const long TSE_O   = HSE_O   + 1167360;
    const long PAIRA_O = TSE_O   + 1167360;          //               (-> HID later)
    const long PAIRB_O = PAIRA_O + 1167360;
    const long PAIRC_O = PAIRB_O + 1167360;
    const long IDX_O   = PAIRC_O + 1167360;          // 2*1520 ints

    float* ENTM  = ws + ENTM_O;
    float* EQ    = ws + EQ_O;
    float* EATT  = ws + EATT_O;
    float* HTATT = ws + HTATT_O;
    float* RS    = ws + RS_O;
    float* HS    = ws + HS_O;
    float* TSb   = ws + TS_O;
    float* HSE   = ws + HSE_O;
    float* TSE   = ws + TSE_O;
    float* PAIRA = ws + PAIRA_O;
    float* PAIRB = ws + PAIRB_O;
    float* PAIRC = ws + PAIRC_O;
    float* HTQ   = HTATT;        // alias: htatt dead after rs GEMM
    float* SC    = ENTM;         // alias: entm dead after k_mention
    float* QB    = HS;           // alias: hs dead after hsE GEMM
    float* KB    = TSb;          // alias: ts dead after tsE GEMM
    float* VB    = RS;           // alias: rs dead after hsE/tsE GEMMs
    float* HID   = PAIRA;        // alias: pairA dead after Wpp GEMM
    int*   HIF   = (int*)(ws + IDX_O);
    int*   TIF   = HIF + BB * NP;

    const int MP = BB * NP;      // 1520 flattened pair rows

    auto gemm = [&](const float* A0, const int* A0i, int lda0, int w0,
                    const float* A1, const int* A1i, int lda1, int w1,
                    const float* A2, const int* A2i, int lda2, int w2,
                    const float* Bm, int ldb, int transB,
                    const float* bias, const float* resid, int ldres,
                    float* Cp, int ldc, int M, int N, int K,
                    float alpha, int act, int batch,
                    long sA0, long sB, long sCst, long sRes) {
        dim3 grid((N + 63) / 64, (M + 63) / 64, batch);
        gemm_wmma_f32<<<grid, 256, 0, stream>>>(A0, A0i, lda0, w0, A1, A1i, lda1, w1,
                                                A2, A2i, lda2, w2, Bm, ldb, transB,
                                                bias, resid, ldres, Cp, ldc, M, N, K,
                                                alpha, act, sA0, sB, sCst, sRes);
    };

    // 1. split hts -> hif/tif
    k_idx<<<(MP + 255) / 256, 256, 0, stream>>>(hts, HIF, TIF, MP);

    // 2. gather masked mention embeddings: entm [B,NE,MM,H]
    k_entm<<<BB * NE * MMN, 256, 0, stream>>>(seq, midx, mmask, ENTM);

    // 3. eq = entm @ Weq + beq   [1840 x 128]  (factored out of per-pair einsum)
    gemm(ENTM, 0, HH, HH, 0,0,0,0, 0,0,0,0, Weq, QSZ, 0, beq, 0,0,
         EQ, QSZ, BB*NE*MMN, QSZ, HH, 1.f, 0, 1, 0,0,0,0);

    // 4. eatt [B,NE,NH,C]
    k_entatt<<<BB * NE * NHD, 256, 0, stream>>>(att, midx, mmask, EATT);

    // 5. htatt [B,P,C]  (head-product pooling + normalize)
    k_htatt<<<MP, 256, 0, stream>>>(EATT, HIF, TIF, HTATT);

    // 6. rs[b] = htatt[b] @ seq[b]   batched [380x512]@[512x768]
    gemm(HTATT, 0, CC, CC, 0,0,0,0, 0,0,0,0, seq, HH, 0, 0, 0,0,
         RS, HH, NP, HH, CC, 1.f, 0, BB,
         (long)NP*CC, (long)CC*HH, (long)NP*HH, 0);

    // 7. htq = rs @ Wcq + bcq   [1520 x 128]
    gemm(RS, 0, HH, HH, 0,0,0,0, 0,0,0,0, Wcq, QSZ, 0, bcq, 0,0,
         HTQ, QSZ, MP, QSZ, HH, 1.f, 0, 1, 0,0,0,0);

    // 8. mention softmax pooling -> hs, ts [1520 x 768]
    k_mention<<<MP, 256, 0, stream>>>(HTQ, EQ, ENTM, mmask, HIF, TIF, HS, TSb);

    // 9. hsE = tanh([hs, rs] @ Whe + bhe);  tsE = tanh([ts, rs] @ Wte + bte)
    gemm(HS, 0, HH, HH, RS, 0, HH, HH, 0,0,0,0, Whe, EMB, 0, bhe, 0,0,
         HSE, EMB, MP, EMB, 2*HH, 1.f, 1, 1, 0,0,0,0);
    gemm(TSb, 0, HH, HH, RS, 0, HH, HH, 0,0,0,0, Wte, EMB, 0, bte, 0,0,
         TSE, EMB, MP, EMB, 2*HH, 1.f, 1, 1, 0,0,0,0);

    // 10. grouped bilinear + Wbl: pairA [1520 x 768]  (dominant GEMM, K=49152)
    {
        dim3 grid(EMB / 64, (MP + 63) / 64, 1);
        gemm_bilinear_wmma<<<grid, 256, 0, stream>>>(HSE, TSE, Wbl, bbl, PAIRA, MP);
    }

    // 11. pairB = tanh([pos[hi], pairA, pos[ti]] @ Wpp + bpp)
    gemm(pos, HIF, POSZ, POSZ, PAIRA, 0, EMB, EMB, pos, TIF, POSZ, POSZ,
         Wpp, EMB, 0, bpp, 0,0, PAIRB, EMB, MP, EMB, POSZ + EMB + POSZ,
         1.f, 1, 1, 0,0,0,0);

    // 12. q/k/v = pairB @ Wg{q,k,v}  (no bias)
    gemm(PAIRB, 0, EMB, EMB, 0,0,0,0, 0,0,0,0, Wgq, EMB, 0, 0, 0,0,
         QB, EMB, MP, EMB, EMB, 1.f, 0, 1, 0,0,0,0);
    gemm(PAIRB, 0, EMB, EMB, 0,0,0,0, 0,0,0,0, Wgk, EMB, 0, 0, 0,0,
         KB, EMB, MP, EMB, EMB, 1.f, 0, 1, 0,0,0,0);
    gemm(PAIRB, 0, EMB, EMB, 0,0,0,0, 0,0,0,0, Wgv, EMB, 0, 0, 0,0,
         VB, EMB, MP, EMB, EMB, 1.f, 0, 1, 0,0,0,0);

    // 13. scores sc[b] = (q[b] @ k[b]^T) / sqrt(EMB)   batched, transB
    gemm(QB, 0, EMB, EMB, 0,0,0,0, 0,0,0,0, KB, EMB, 1, 0, 0,0,
         SC, NP, NP, NP, EMB, INV_SQRT_EMB, 0, BB,
         (long)NP*EMB, (long)NP*EMB, (long)NP*NP, 0);

    // 14. masked softmax over pairs, in place
    k_pairsoftmax<<<MP, 256, 0, stream>>>(SC, vis);

    // 15. pairC = pairB + softmax(sc) @ v   batched, residual epilogue
    gemm(SC, 0, NP, NP, 0,0,0,0, 0,0,0,0, VB, EMB, 0, 0, PAIRB, EMB,
         PAIRC, EMB, NP, EMB, NP, 1.f, 0, BB,
         (long)NP*NP, (long)NP*EMB, (long)NP*EMB, (long)NP*EMB);

    // 16. hid = tanh([hsE, tsE, pairC] @ Wp1 + bp1)
    gemm(HSE, 0, EMB, EMB, TSE, 0, EMB, EMB, PAIRC, 0, EMB, EMB,
         Wp1, EMB, 0, bp1, 0,0, HID, EMB, MP, EMB, 3*EMB, 1.f, 1, 1, 0,0,0,0);

    // 17. logits = hid @ Wp2 + bp2  -> d_out [1520 x 97]
    gemm(HID, 0, EMB, EMB, 0,0,0,0, 0,0,0,0, Wp2, NCLS, 0, bp2, 0,0,
         out, NCLS, MP, NCLS, EMB, 1.f, 0, 1, 0,0,0,0);
}